// Propagate_40381282517566
// MI455X (gfx1250) — compile-verified
//
#include <hip/hip_runtime.h>
#include <hip/hip_bf16.h>
#include <stdint.h>

#define D_FEAT        128
#define E_CHUNK       512        // edges staged per chunk (per block)
#define WAVES         8
#define BATCH         16         // rows per TDM gather (16-bit index mode max)
#define SCATTER_BLOCKS 512

typedef unsigned int u32x4 __attribute__((ext_vector_type(4)));
typedef int          i32x8 __attribute__((ext_vector_type(8)));
typedef int          i32x4 __attribute__((ext_vector_type(4)));

#if defined(__has_builtin)
#if __has_builtin(__builtin_amdgcn_s_wait_tensorcnt)
#define WAIT_TENSOR(n) __builtin_amdgcn_s_wait_tensorcnt((short)(n))
#endif
#endif
#ifndef WAIT_TENSOR
#define WAIT_TENSOR(n) asm volatile("s_wait_tensorcnt %0" :: "n"(n) : "memory")
#endif

// ---------------------------------------------------------------------------
// TDM gather: fetch `bcnt` rows (<=16) of the 2D tensor Ybase[n_nodes][128]
// into LDS at byte offset lds_addr, row indices taken from eidx[0..bcnt).
// D# layout per CDNA5 ISA ch.8 (gather mode, 16-bit indices, f32 elements).
// ---------------------------------------------------------------------------
__device__ __forceinline__ void tdm_gather_rows(const float* Ybase,
                                                unsigned lds_addr,
                                                const int* eidx, int bcnt,
                                                int n_nodes) {
    unsigned long long ga = (unsigned long long)(size_t)Ybase;
    u32x4 g0;
    g0.x = 0x80000001u;                                   // gather_mode=1, idx16, count=1
    g0.y = lds_addr;                                      // LDS dest (bytes)
    g0.z = (unsigned)(ga & 0xffffffffu);                  // global_addr[31:0]
    g0.w = (unsigned)((ga >> 32) & 0x01ffffffu) | (2u << 30); // addr[56:32] | type=2

    i32x8 g1;
    g1[0] = (2 << 16);                                    // data_size=4B, wg_mask=0
    g1[1] = (int)((unsigned)(D_FEAT & 0xffff) << 16);     // tensor_dim0 lo16
    g1[2] = (int)(((unsigned)D_FEAT >> 16) |
                  (((unsigned)n_nodes & 0xffffu) << 16)); // tensor_dim0 hi | tensor_dim1 lo
    g1[3] = (int)((((unsigned)n_nodes >> 16) & 0xffffu) |
                  ((unsigned)D_FEAT << 16));              // tensor_dim1 hi | tile_dim0=128
    g1[4] = bcnt;                                         // tile_dim1 = #indices, tile_dim2=0
    g1[5] = D_FEAT;                                       // tensor_dim0_stride lo32 (elements)
    g1[6] = 0;
    g1[7] = 0;

    i32x4 g2, g3;
#pragma unroll
    for (int k = 0; k < 4; ++k) {
        int a = 2 * k, b = 2 * k + 1;
        int i0 = (a < bcnt) ? eidx[a] : 0;
        int i1 = (b < bcnt) ? eidx[b] : 0;
        g2[k] = (i0 & 0xffff) | (i1 << 16);
    }
#pragma unroll
    for (int k = 0; k < 4; ++k) {
        int a = 8 + 2 * k, b = 9 + 2 * k;
        int i0 = (a < bcnt) ? eidx[a] : 0;
        int i1 = (b < bcnt) ? eidx[b] : 0;
        g3[k] = (i0 & 0xffff) | (i1 << 16);
    }

#if defined(__has_builtin) && __has_builtin(__builtin_amdgcn_tensor_load_to_lds)
#if __clang_major__ >= 23
    // 6-arg (therock/clang-23) form: extra int32x8 operand; mirror groups 2+3
    // into it so the index data is correct whichever slot the intrinsic uses.
    i32x8 g23;
    g23[0] = g2[0]; g23[1] = g2[1]; g23[2] = g2[2]; g23[3] = g2[3];
    g23[4] = g3[0]; g23[5] = g3[1]; g23[6] = g3[2]; g23[7] = g3[3];
    __builtin_amdgcn_tensor_load_to_lds(g0, g1, g2, g3, g23, 0);
#else
    // 5-arg (ROCm 7.2 / clang-22) form.
    __builtin_amdgcn_tensor_load_to_lds(g0, g1, g2, g3, 0);
#endif
#else
    asm volatile("tensor_load_to_lds %0, %1, %2, %3"
                 :: "s"(g0), "s"(g1), "s"(g2), "s"(g3) : "memory");
#endif
}

// ---------------------------------------------------------------------------
// 1) zero workspace (deg + Zacc) — must run every call (graph replay)
// ---------------------------------------------------------------------------
__global__ void k_zero(int* __restrict__ deg, float* __restrict__ zacc,
                       int n_nodes) {
    long stride = (long)gridDim.x * blockDim.x;
    long i0 = (long)blockIdx.x * blockDim.x + threadIdx.x;
    for (long i = i0; i < n_nodes; i += stride) deg[i] = 0;
    long total = (long)n_nodes * D_FEAT;
    for (long i = i0; i < total; i += stride) zacc[i] = 0.0f;
}

// ---------------------------------------------------------------------------
// 2) in-degree histogram (global_atomic_add_u32, non-returning)
// ---------------------------------------------------------------------------
__global__ void k_deg(const int* __restrict__ dst, int* __restrict__ deg,
                      int n_edges) {
    int e = blockIdx.x * blockDim.x + threadIdx.x;
    if (e < n_edges) atomicAdd(&deg[dst[e]], 1);
}

// ---------------------------------------------------------------------------
// 3) s1[n] = (lam*deg + (1-lam))^(-1/2)
// ---------------------------------------------------------------------------
__global__ void k_scale(const int* __restrict__ deg,
                        const float* __restrict__ lam_p,
                        float* __restrict__ s1, int n_nodes) {
    float lam = *lam_p;
    int n = blockIdx.x * blockDim.x + threadIdx.x;
    if (n < n_nodes) {
        float t = lam * (float)deg[n] + (1.0f - lam);
        s1[n] = rsqrtf(t);
    }
}

// ---------------------------------------------------------------------------
// 4) edge scatter: Zacc[d] += s1[s] * Y[s]
//    - edge indices streamed into LDS via async global->LDS (ASYNCcnt),
//      double-buffered across chunks;
//    - Y rows gathered 16-at-a-time by the Tensor Data Mover in gather mode
//      (TENSORcnt), double-buffered per wave against the atomic scatter.
// ---------------------------------------------------------------------------
__global__ __launch_bounds__(256) void k_scatter(
    const float* __restrict__ Y, const int* __restrict__ src,
    const int* __restrict__ dst, const float* __restrict__ s1,
    float* __restrict__ zacc, int n_edges, int n_nodes) {

    __shared__ int   lds_src[2][E_CHUNK];
    __shared__ int   lds_dst[2][E_CHUNK];
    __shared__ float rowbuf[WAVES][2][BATCH * D_FEAT];   // 8KB per wave per buffer

    const int tid  = threadIdx.x;
    const int lane = tid & 31;
    const int wave = tid >> 5;                 // 0..7
    const int t2   = tid * 2;                  // 2 ints (8B) staged per thread per array
    const int n_chunks = (n_edges + E_CHUNK - 1) / E_CHUNK;
    const unsigned gmax = (unsigned)(n_edges - 2) * 4u;   // clamp: 8B load in-bounds

    auto issue = [&](int c, int b) {
        unsigned goff = ((unsigned)c * E_CHUNK + (unsigned)t2) * 4u;
        if (goff > gmax) goff = gmax;          // tail: load garbage we never read
        unsigned lsrc = (unsigned)(size_t)&lds_src[b][t2];
        unsigned ldst = (unsigned)(size_t)&lds_dst[b][t2];
        asm volatile("global_load_async_to_lds_b64 %0, %1, %2"
                     :: "v"(lsrc), "v"(goff), "s"(src) : "memory");
        asm volatile("global_load_async_to_lds_b64 %0, %1, %2"
                     :: "v"(ldst), "v"(goff), "s"(dst) : "memory");
    };

    const int c0 = blockIdx.x;
    if (c0 < n_chunks) issue(c0, 0);

    const unsigned lbuf0 = (unsigned)(size_t)&rowbuf[wave][0][0];
    const unsigned lbuf1 = (unsigned)(size_t)&rowbuf[wave][1][0];

    int buf = 0;
    for (int c = c0; c < n_chunks; c += gridDim.x, buf ^= 1) {
        const int cn = c + gridDim.x;
        if (cn < n_chunks) {
            issue(cn, buf ^ 1);                          // prefetch next edge chunk
            asm volatile("s_wait_asynccnt 2" ::: "memory");
        } else {
            asm volatile("s_wait_asynccnt 0" ::: "memory");
        }
        __syncthreads();                                 // staged edges visible

        int cnt = n_edges - c * E_CHUNK;
        if (cnt > E_CHUNK) cnt = E_CHUNK;
        const int ebase = wave * (E_CHUNK / WAVES);      // 64 edges per wave
        int ecnt = cnt - ebase;
        if (ecnt > E_CHUNK / WAVES) ecnt = E_CHUNK / WAVES;
        const int nb = (ecnt > 0) ? (ecnt + BATCH - 1) / BATCH : 0;

        if (nb > 0) {
            int b0cnt = (ecnt < BATCH) ? ecnt : BATCH;
            tdm_gather_rows(Y, lbuf0, &lds_src[buf][ebase], b0cnt, n_nodes);
        }
        for (int bI = 0; bI < nb; ++bI) {
            const int b0 = bI * BATCH;
            int bcnt = ecnt - b0;
            if (bcnt > BATCH) bcnt = BATCH;
            if (bI + 1 < nb) {
                int nc = ecnt - b0 - BATCH;
                if (nc > BATCH) nc = BATCH;
                tdm_gather_rows(Y, ((bI + 1) & 1) ? lbuf1 : lbuf0,
                                &lds_src[buf][ebase + b0 + BATCH], nc, n_nodes);
                WAIT_TENSOR(1);                          // current batch landed
            } else {
                WAIT_TENSOR(0);
            }
            asm volatile("" ::: "memory");               // order LDS reads after wait

            const float* rb = &rowbuf[wave][bI & 1][0];
            for (int j = 0; j < bcnt; ++j) {
                const int s = lds_src[buf][ebase + b0 + j];
                const int d = lds_dst[buf][ebase + b0 + j];
                const float sc = s1[s];                  // L2-resident (200 KB)
                const float4 v = *(const float4*)(rb + j * D_FEAT + lane * 4);
                float* zp = zacc + (long)d * D_FEAT + lane * 4;
                atomicAdd(zp + 0, v.x * sc);             // global_atomic_add_f32
                atomicAdd(zp + 1, v.y * sc);
                atomicAdd(zp + 2, v.z * sc);
                atomicAdd(zp + 3, v.w * sc);
            }
        }
        __syncthreads();                                 // done reading buf
    }
}

// ---------------------------------------------------------------------------
// 5) out = (1-alp)*Y + (alp*lam*s1[n])*Zacc + (alp*s1[n]^2)*X
// ---------------------------------------------------------------------------
__global__ void k_combine(const float* __restrict__ Y,
                          const float* __restrict__ X,
                          const float* __restrict__ alp_p,
                          const float* __restrict__ lam_p,
                          const float* __restrict__ s1,
                          const float* __restrict__ zacc,
                          float* __restrict__ out, int n_nodes) {
    const float alp = *alp_p;
    const float lam = *lam_p;
    const long nvec = (long)n_nodes * (D_FEAT / 4);
    long i = (long)blockIdx.x * blockDim.x + threadIdx.x;
    if (i >= nvec) return;
    const int n = (int)(i >> 5);               // 32 float4 per 128-wide row
    const float s  = s1[n];
    const float cy = 1.0f - alp;
    const float cz = alp * lam * s;
    const float cx = alp * s * s;
    const float4 y = ((const float4*)Y)[i];
    const float4 x = ((const float4*)X)[i];
    const float4 z = ((const float4*)zacc)[i];
    float4 o;
    o.x = cy * y.x + cz * z.x + cx * x.x;
    o.y = cy * y.y + cz * z.y + cx * x.y;
    o.z = cy * y.z + cz * z.z + cx * x.z;
    o.w = cy * y.w + cz * z.w + cx * x.w;
    ((float4*)out)[i] = o;
}

// ---------------------------------------------------------------------------
// launcher
// ---------------------------------------------------------------------------
extern "C" void kernel_launch(void* const* d_in, const int* in_sizes, int n_in,
                              void* d_out, int out_size, void* d_ws, size_t ws_size,
                              hipStream_t stream) {
    const float* Y     = (const float*)d_in[0];
    const float* X     = (const float*)d_in[1];
    const float* alp_p = (const float*)d_in[2];
    const float* lam_p = (const float*)d_in[3];
    const int*   src   = (const int*)d_in[4];
    const int*   dst   = (const int*)d_in[5];

    const int n_nodes = in_sizes[0] / D_FEAT;
    const int n_edges = in_sizes[4];

    // ws layout: [deg: n_nodes int][s1: n_nodes float][Zacc: n_nodes*128 float]
    int*   deg  = (int*)d_ws;
    float* s1   = (float*)d_ws + n_nodes;
    long   zoff = ((long)(2 * n_nodes) + 3) & ~3L;   // 16B-align Zacc for float4
    float* zacc = (float*)d_ws + zoff;

    k_zero<<<2048, 256, 0, stream>>>(deg, zacc, n_nodes);
    k_deg<<<(n_edges + 255) / 256, 256, 0, stream>>>(dst, deg, n_edges);
    k_scale<<<(n_nodes + 255) / 256, 256, 0, stream>>>(deg, lam_p, s1, n_nodes);
    k_scatter<<<SCATTER_BLOCKS, 256, 0, stream>>>(Y, src, dst, s1, zacc,
                                                  n_edges, n_nodes);

    const long nvec = (long)n_nodes * (D_FEAT / 4);
    k_combine<<<(int)((nvec + 255) / 256), 256, 0, stream>>>(
        Y, X, alp_p, lam_p, s1, zacc, (float*)d_out, n_nodes);
}